// subspace_linop_22170621182458
// MI455X (gfx1250) — compile-verified
//
#include <hip/hip_runtime.h>

typedef float v2f __attribute__((ext_vector_type(2)));
typedef float v8f __attribute__((ext_vector_type(8)));

#define NCOIL 12
#define NSUB  5
#define NIMG  (NCOIL * NSUB)   // 60 complex planes
#define NX    256
#define NY    256
#define PLANE (NX * NY)        // 65536
#define NRO   960
#define NPE   32
#define NTR   32
#define NPTS  (NRO * NPE * NTR) // 983040

// ---------------------------------------------------------------------------
// Twiddle table: W[k][n] = exp(-2*pi*i * k*n / 256), exact via (k*n mod 256).
// ---------------------------------------------------------------------------
__global__ void __launch_bounds__(256) build_twiddle(float* __restrict__ Wr,
                                                     float* __restrict__ Wi) {
    int idx = blockIdx.x * 256 + threadIdx.x;   // idx = k*256 + n
    int k = idx >> 8;
    int n = idx & 255;
    int m = (k * n) & 255;                      // periodic phase
    float th = (float)m * 0.024543692606170259f; // 2*pi/256
    float s, c;
    __sincosf(th, &s, &c);
    Wr[idx] = c;
    Wi[idx] = -s;
}

// ---------------------------------------------------------------------------
// fp32 WMMA wrapper: D = A(16x4) * B(4x16) + C(16x16), wave32.
// ---------------------------------------------------------------------------
__device__ __forceinline__ v8f wmma_f32(v2f a, v2f b, v8f c) {
    return __builtin_amdgcn_wmma_f32_16x16x4_f32(
        /*neg_a=*/false, a, /*neg_b=*/false, b,
        /*c_mod=*/(short)0, c, /*reuse_a=*/false, /*reuse_b=*/false);
}

// ---------------------------------------------------------------------------
// Pass 1 (y-transform): G[img][x][ky] = sum_ny (mps*alphas)[x][ny] * W[ky][ny]
// One workgroup = one (img, 16-row strip). Phase 1 computes S = mps*alphas for
// the strip ONCE into LDS (split re/im). Phase 2: 8 waves x 2 N-tiles each,
// A from LDS (ds_load), B (twiddles) from global/L2.
// ---------------------------------------------------------------------------
__global__ void __launch_bounds__(256) dft_pass_y(
    const float2* __restrict__ mps, const float2* __restrict__ alphas,
    const float* __restrict__ Wr, const float* __restrict__ Wi,
    float2* __restrict__ G) {
    __shared__ float ldsSr[16 * 256];
    __shared__ float ldsSi[16 * 256];

    const int img = blockIdx.x >> 4;            // 0..59
    const int tm  = (blockIdx.x & 15) << 4;     // strip row base
    const int c   = img / NSUB;
    const int s   = img % NSUB;
    const float2* mp = mps    + c * PLANE;
    const float2* al = alphas + s * PLANE;

    // ---- Phase 1: strip S into LDS (2048 complex pairs, 8 pairs/thread) ----
    const int tid = threadIdx.x;
#pragma unroll
    for (int j = 0; j < 8; j++) {
        const int q   = tid + 256 * j;          // pair index (2 complexes)
        const int row = q >> 7;                 // local row 0..15
        const int kk  = (q << 1) & 255;         // column 0,2,4,...
        float4 m4 = *reinterpret_cast<const float4*>(mp + (tm + row) * NY + kk);
        float4 a4 = *reinterpret_cast<const float4*>(al + (tm + row) * NY + kk);
        float2 sr, si;
        sr.x = m4.x * a4.x - m4.y * a4.y;
        si.x = m4.x * a4.y + m4.y * a4.x;
        sr.y = m4.z * a4.z - m4.w * a4.w;
        si.y = m4.z * a4.w + m4.w * a4.z;
        *reinterpret_cast<float2*>(&ldsSr[row * 256 + kk]) = sr;
        *reinterpret_cast<float2*>(&ldsSi[row * 256 + kk]) = si;
    }
    __syncthreads();

    // ---- Phase 2: WMMA, 2 column-tiles per wave ----
    const int wave = threadIdx.x >> 5;
    const int lane = threadIdx.x & 31;
    const int l15  = lane & 15;
    const int koff = (lane >> 4) << 1;          // lanes 16-31 hold K+2,K+3
    const int tn0  = wave << 5;                 // col tile 0 base
    const int col0 = tn0 + l15;
    const int col1 = col0 + 16;

    v8f accR0 = {}, accI0 = {}, accR1 = {}, accI1 = {};

    for (int k = 0; k < NY; k += 4) {
        const int kk = k + koff;
        v2f Ar = *reinterpret_cast<const v2f*>(&ldsSr[l15 * 256 + kk]);
        v2f Ai = *reinterpret_cast<const v2f*>(&ldsSi[l15 * 256 + kk]);

        float2 br0 = *reinterpret_cast<const float2*>(Wr + col0 * NY + kk);
        float2 bi0 = *reinterpret_cast<const float2*>(Wi + col0 * NY + kk);
        float2 br1 = *reinterpret_cast<const float2*>(Wr + col1 * NY + kk);
        float2 bi1 = *reinterpret_cast<const float2*>(Wi + col1 * NY + kk);
        v2f Br0 = { br0.x,  br0.y}, Bi0 = { bi0.x,  bi0.y}, Bn0 = {-bi0.x, -bi0.y};
        v2f Br1 = { br1.x,  br1.y}, Bi1 = { bi1.x,  bi1.y}, Bn1 = {-bi1.x, -bi1.y};

        accR0 = wmma_f32(Ar, Br0, accR0);
        accR0 = wmma_f32(Ai, Bn0, accR0);
        accI0 = wmma_f32(Ar, Bi0, accI0);
        accI0 = wmma_f32(Ai, Br0, accI0);
        accR1 = wmma_f32(Ar, Br1, accR1);
        accR1 = wmma_f32(Ai, Bn1, accR1);
        accI1 = wmma_f32(Ar, Bi1, accI1);
        accI1 = wmma_f32(Ai, Br1, accI1);
    }

    float2* g = G + img * PLANE;
    const int rbase = tm + ((lane >> 4) << 3);
#pragma unroll
    for (int v = 0; v < 8; v++) {
        g[(rbase + v) * NY + tn0 + l15]      = make_float2(accR0[v], accI0[v]);
        g[(rbase + v) * NY + tn0 + 16 + l15] = make_float2(accR1[v], accI1[v]);
    }
}

// ---------------------------------------------------------------------------
// Pass 2 (x-transform): F[img][kx][ky] = sum_x W[kx][x] * G[img][x][ky]
// One workgroup = one (img, 16-col ky strip). Phase 1 stages the G strip into
// LDS *transposed* via async global->LDS (per-lane LDS scatter), layout
// ldsG[col][x] (interleaved re/im). Phase 2: 8 waves x 2 kx-tiles each,
// A (twiddles) from global/L2, B from LDS as one b128 per K-step.
// ---------------------------------------------------------------------------
__global__ void __launch_bounds__(256) dft_pass_x(
    const float* __restrict__ Wr, const float* __restrict__ Wi,
    const float2* __restrict__ G, float2* __restrict__ F) {
    __shared__ float2 ldsG[16 * 256];           // [col][x], 32 KB

    const int img = blockIdx.x >> 4;
    const int tn  = (blockIdx.x & 15) << 4;     // ky strip base
    const float2* g = G + img * PLANE;

    // ---- Phase 1: async copy strip (4096 complex, 16/thread), transposing ----
    const int tid = threadIdx.x;
#pragma unroll
    for (int j = 0; j < 16; j++) {
        const int e   = tid + 256 * j;          // 0..4095
        const int col = e & 15;                 // ky - tn
        const int row = e >> 4;                 // x
        unsigned lds_off = (unsigned)(size_t)(&ldsG[col * 256 + row]);
        const float2* gp = g + row * NY + tn + col;
        asm volatile("global_load_async_to_lds_b64 %0, %1, off"
                     :: "v"(lds_off), "v"(gp) : "memory");
    }
    asm volatile("s_wait_asynccnt 0" ::: "memory");
    __syncthreads();

    // ---- Phase 2: WMMA, 2 row-tiles per wave ----
    const int wave = threadIdx.x >> 5;
    const int lane = threadIdx.x & 31;
    const int l15  = lane & 15;
    const int koff = (lane >> 4) << 1;
    const int row0 = (wave << 5) + l15;         // kx tile 0 row
    const int row1 = row0 + 16;

    v8f accR0 = {}, accI0 = {}, accR1 = {}, accI1 = {};

    for (int k = 0; k < NX; k += 4) {
        const int kk = k + koff;
        float2 a0r = *reinterpret_cast<const float2*>(Wr + row0 * NX + kk);
        float2 a0i = *reinterpret_cast<const float2*>(Wi + row0 * NX + kk);
        float2 a1r = *reinterpret_cast<const float2*>(Wr + row1 * NX + kk);
        float2 a1i = *reinterpret_cast<const float2*>(Wi + row1 * NX + kk);
        v2f Ar0 = {a0r.x, a0r.y}, Ai0 = {a0i.x, a0i.y};
        v2f Ar1 = {a1r.x, a1r.y}, Ai1 = {a1i.x, a1i.y};

        // B: ldsG[l15][kk], ldsG[l15][kk+1] -> (gr0, gi0, gr1, gi1)
        float4 bq = *reinterpret_cast<const float4*>(&ldsG[l15 * 256 + kk]);
        v2f Br = {bq.x, bq.z}, Bi = {bq.y, bq.w}, Bn = {-bq.y, -bq.w};

        accR0 = wmma_f32(Ar0, Br, accR0);
        accR0 = wmma_f32(Ai0, Bn, accR0);
        accI0 = wmma_f32(Ar0, Bi, accI0);
        accI0 = wmma_f32(Ai0, Br, accI0);
        accR1 = wmma_f32(Ar1, Br, accR1);
        accR1 = wmma_f32(Ai1, Bn, accR1);
        accI1 = wmma_f32(Ar1, Bi, accI1);
        accI1 = wmma_f32(Ai1, Br, accI1);
    }

    float2* f = F + img * PLANE;
    const int colD = tn + l15;
    const int roff = (lane >> 4) << 3;
#pragma unroll
    for (int v = 0; v < 8; v++) {
        f[((wave << 5) + roff + v) * NY + colD]      = make_float2(accR0[v], accI0[v]);
        f[((wave << 5) + 16 + roff + v) * NY + colD] = make_float2(accR1[v], accI1[v]);
    }
}

// ---------------------------------------------------------------------------
// Gather at rounded trajectory points + phi projection over nsub.
// sign = (-1)^(ix+iy) implements fft(ifftshift(.)) for even N.
// out[c,r,pe,t] = sign * sum_s phi[s][t] * F[c,s][ix*NY+iy]
// ---------------------------------------------------------------------------
__global__ void __launch_bounds__(256) gather_phi(
    const float* __restrict__ trj, const float2* __restrict__ F,
    const float2* __restrict__ phi, float2* __restrict__ out) {
    const int p = blockIdx.x * 256 + threadIdx.x;  // (r*NPE + pe)*NTR + t
    const int t = p & (NTR - 1);

    float2 tr = reinterpret_cast<const float2*>(trj)[p];
    int ix = __float2int_rn(tr.x) & 255;   // two's-complement & == nonneg mod 256
    int iy = __float2int_rn(tr.y) & 255;
    const float sgn = ((ix + iy) & 1) ? -1.0f : 1.0f;
    const int idx = ix * NY + iy;

    float2 ph[NSUB];
#pragma unroll
    for (int s = 0; s < NSUB; s++) ph[s] = phi[s * NTR + t];

#pragma unroll
    for (int c = 0; c < NCOIL; c++) {
        float ar = 0.0f, ai = 0.0f;
#pragma unroll
        for (int s = 0; s < NSUB; s++) {
            float2 f = F[(c * NSUB + s) * PLANE + idx];
            ar += ph[s].x * f.x - ph[s].y * f.y;
            ai += ph[s].x * f.y + ph[s].y * f.x;
        }
        out[c * NPTS + p] = make_float2(sgn * ar, sgn * ai);
    }
}

// ---------------------------------------------------------------------------
// Launch
// ---------------------------------------------------------------------------
extern "C" void kernel_launch(void* const* d_in, const int* in_sizes, int n_in,
                              void* d_out, int out_size, void* d_ws, size_t ws_size,
                              hipStream_t stream) {
    const float2* alphas = (const float2*)d_in[0];  // (5,256,256) complex64
    const float2* mps    = (const float2*)d_in[1];  // (12,256,256) complex64
    const float2* phi    = (const float2*)d_in[2];  // (5,32) complex64
    const float*  trj    = (const float*) d_in[3];  // (960,32,32,2) float32

    float*  ws = (float*)d_ws;
    float*  Wr = ws;                                // 64K floats
    float*  Wi = ws + PLANE;                        // 64K floats
    float2* G  = (float2*)(ws + 2 * PLANE);         // 60 * 64K complex
    float2* F  = G + (size_t)NIMG * PLANE;          // 60 * 64K complex
    float2* out = (float2*)d_out;                   // (12,960,32,32) complex64

    build_twiddle<<<PLANE / 256, 256, 0, stream>>>(Wr, Wi);
    dft_pass_y  <<<NIMG * 16, 256, 0, stream>>>(mps, alphas, Wr, Wi, G);
    dft_pass_x  <<<NIMG * 16, 256, 0, stream>>>(Wr, Wi, G, F);
    gather_phi  <<<NPTS / 256, 256, 0, stream>>>(trj, F, phi, out);
}